// RPNTrainer_42494406427387
// MI455X (gfx1250) — compile-verified
//
#include <hip/hip_runtime.h>
#include <hip/hip_bf16.h>
#include <math.h>

#define A_N     36864
#define B_N     8
#define T_N     64
#define TOT_N   (A_N * B_N)      // 294912
#define POS_CAP 128
#define NEG_CAP 248              // 256 - B

#define AS1 __attribute__((address_space(1)))
#define AS3 __attribute__((address_space(3)))

typedef __attribute__((ext_vector_type(2))) float v2f;
typedef __attribute__((ext_vector_type(8))) float v8f;
typedef __attribute__((ext_vector_type(4))) int   v4i;

// ---------------------------------------------------------------------------
// Wave32 sum routed through the matrix engine.
// D = ones(16x4) x B(4x16) + 0  =>  every element of D is a column-sum of B.
// VGPR0 of D holds two full rows, so sum(c[0] over 32 lanes) == 2 * sum(B).
// This identity holds for any row/lane striping, so it is layout-robust.
// ---------------------------------------------------------------------------
__device__ __forceinline__ float wave_sum32(float v) {
#if __has_builtin(__builtin_amdgcn_wmma_f32_16x16x4_f32)
  v2f a; a[0] = 1.0f; a[1] = 1.0f;      // all-ones A matrix (any layout)
  v2f b; b[0] = v;    b[1] = 0.0f;      // 64-slot B matrix: one value per lane
  v8f c = {};
  c = __builtin_amdgcn_wmma_f32_16x16x4_f32(false, a, false, b, (short)0, c,
                                            false, false);
  float p = c[0];
#pragma unroll
  for (int off = 16; off > 0; off >>= 1) p += __shfl_xor(p, off);
  return 0.5f * p;
#else
#pragma unroll
  for (int off = 16; off > 0; off >>= 1) v += __shfl_xor(v, off);
  return v;
#endif
}

// ---------------------------------------------------------------------------
// Pass 1: per (b, anchor) max-IoU over 64 targets. Targets staged in LDS via
// gfx1250 async-to-LDS. Output: packed byte (bit7 = mask, bits5:0 = argmax t),
// plus global positive count (integer atomics -> deterministic).
// ---------------------------------------------------------------------------
__global__ void iou_kernel(const float* __restrict__ anchors,
                           const float* __restrict__ targets,
                           unsigned char* __restrict__ mb,
                           int* __restrict__ nPos) {
  __shared__ __align__(16) float sT[T_N * 4];
  __shared__ float sArea[T_N];
  __shared__ int sCnt;
  const int tid = threadIdx.x;
  const int b   = blockIdx.y;
  const int a   = blockIdx.x * 256 + tid;
  if (tid == 0) sCnt = 0;

#if __has_builtin(__builtin_amdgcn_global_load_async_to_lds_b128)
  if (tid < T_N) {
    AS1 v4i* g = (AS1 v4i*)(targets + (size_t)(b * T_N + tid) * 4);
    AS3 v4i* l = (AS3 v4i*)(&sT[tid * 4]);
    __builtin_amdgcn_global_load_async_to_lds_b128(g, l, 0, 0);
  }
#if __has_builtin(__builtin_amdgcn_s_wait_asynccnt)
  __builtin_amdgcn_s_wait_asynccnt(0);
#else
  asm volatile("s_wait_asynccnt 0" ::: "memory");
#endif
#else
  if (tid < T_N)
    ((float4*)sT)[tid] = ((const float4*)targets)[b * T_N + tid];
#endif
  __syncthreads();

  if (tid < T_N) {
    float x1 = sT[tid * 4 + 0], y1 = sT[tid * 4 + 1];
    float x2 = sT[tid * 4 + 2], y2 = sT[tid * 4 + 3];
    sArea[tid] = (x2 - x1) * (y2 - y1);
  }
  __syncthreads();

  const float4 an   = ((const float4*)anchors)[a];
  const float areaA = (an.z - an.x) * (an.w - an.y);
  float best = -1.0f;
  int   bi   = 0;
#pragma unroll 4
  for (int t = 0; t < T_N; ++t) {
    const float tx1 = sT[t * 4 + 0], ty1 = sT[t * 4 + 1];
    const float tx2 = sT[t * 4 + 2], ty2 = sT[t * 4 + 3];
    const float ix1 = fmaxf(an.x, tx1), iy1 = fmaxf(an.y, ty1);
    const float ix2 = fminf(an.z, tx2), iy2 = fminf(an.w, ty2);
    const float iw = fmaxf(ix2 - ix1, 0.0f), ih = fmaxf(iy2 - iy1, 0.0f);
    const float inter = iw * ih;
    const float iou = inter / (areaA + sArea[t] - inter);
    if (iou > best) { best = iou; bi = t; }   // strict '>' keeps first max
  }
  const int i = b * A_N + a;
  const int m = (best > 0.5f) ? 1 : 0;
  mb[i] = (unsigned char)(bi | (m << 7));

  const unsigned int bal = (unsigned int)__ballot(m);
  if ((tid & 31) == 0) atomicAdd(&sCnt, __popc(bal));
  __syncthreads();
  if (tid == 0) atomicAdd(nPos, sCnt);
}

// ---------------------------------------------------------------------------
// Pass 2: stable selection. Single 1024-thread block scans flat indices in
// order; pos ordinal via wave32 ballots + cross-wave scan; neg ordinal is
// i - positives_before(i). Early-exits once both caps are satisfied.
// ---------------------------------------------------------------------------
__global__ void select_kernel(const unsigned char* __restrict__ mb,
                              const int* __restrict__ nPosPtr,
                              int* __restrict__ posIdx,
                              int* __restrict__ negIdx) {
  __shared__ int wcnt[32];
  __shared__ int wscan[32];
  __shared__ int sBase, sTot, sDone;
  const int tid  = threadIdx.x;
  const int lane = tid & 31;
  const int w    = tid >> 5;
  if (tid == 0) { sBase = 0; sDone = 0; }
  __syncthreads();
  const int nPos   = *nPosPtr;
  const int posCap = nPos < POS_CAP ? nPos : POS_CAP;

  for (int c = 0; c < TOT_N / 1024; ++c) {       // 288 chunks, exact
    const int i = c * 1024 + tid;
    const int m = (mb[i] >> 7) & 1;
    const unsigned int bal = (unsigned int)__ballot(m);
    const int lanePre = __popc(bal & ((1u << lane) - 1u));
    if (lane == 0) wcnt[w] = __popc(bal);
    __syncthreads();
    if (tid == 0) {
      int s = 0;
      for (int k = 0; k < 32; ++k) { wscan[k] = s; s += wcnt[k]; }
      sTot = s;
    }
    __syncthreads();
    const int base   = sBase;
    const int posPre = base + wscan[w] + lanePre;  // positives in [0, i)
    if (m) {
      if (posPre < POS_CAP) posIdx[posPre] = i;
    } else {
      const int negOrd = i - posPre;               // negatives in [0, i)
      if (negOrd < NEG_CAP) negIdx[negOrd] = i;
    }
    __syncthreads();
    if (tid == 0) {
      const int nb = base + sTot;
      sBase = nb;
      const int processed = (c + 1) * 1024;
      sDone = (nb >= posCap && (processed - nb) >= NEG_CAP) ? 1 : 0;
    }
    __syncthreads();
    if (sDone) break;                               // uniform
  }
}

// ---------------------------------------------------------------------------
// Pass 3: BCE over 376 selected logits + Huber over <=128 positives.
// Reductions finished through the matrix engine (wave_sum32).
// ---------------------------------------------------------------------------
__global__ void loss_kernel(const float* __restrict__ reg,
                            const float* __restrict__ cls,
                            const float* __restrict__ anchors,
                            const float* __restrict__ targets,
                            const unsigned char* __restrict__ mb,
                            const int* __restrict__ posIdx,
                            const int* __restrict__ negIdx,
                            const int* __restrict__ nPosPtr,
                            float* __restrict__ out) {
  __shared__ float sB[8], sH[8];
  const int tid  = threadIdx.x;
  const int lane = tid & 31;
  const int w    = tid >> 5;
  const int nPos = *nPosPtr;
  const int posV = nPos < POS_CAP ? nPos : POS_CAP;
  const int negA = TOT_N - nPos;
  const int negV = negA < NEG_CAP ? negA : NEG_CAP;

  float bce = 0.0f;
  for (int s = tid; s < POS_CAP + NEG_CAP; s += 256) {
    int idx; float label; bool valid;
    if (s < POS_CAP) { idx = posIdx[s];           label = 1.0f; valid = (s < posV); }
    else             { idx = negIdx[s - POS_CAP]; label = 0.0f; valid = ((s - POS_CAP) < negV); }
    if (valid) {
      const float x  = cls[idx];
      const float sp = fmaxf(x, 0.0f) + log1pf(expf(-fabsf(x)));  // softplus
      bce += sp - x * label;
    }
  }

  float hub = 0.0f;
  if (tid < posV) {
    const int i = posIdx[tid];
    const int b = i / A_N;
    const int a = i - b * A_N;
    const int t = mb[i] & 63;
    const float* tb = targets + ((size_t)(b * T_N + t) << 2);
    const float* ab = anchors + ((size_t)a << 2);
    const float* rb = reg + ((size_t)i << 2);
#pragma unroll
    for (int cN = 0; cN < 4; ++cN) {
      const float d  = rb[cN] - (tb[cN] - ab[cN]);
      const float ad = fabsf(d);
      hub += (ad < 1.0f) ? 0.5f * d * d : ad - 0.5f;
    }
  }

  const float wb = wave_sum32(bce);   // all 256 threads reach here: EXEC all-1s
  const float wh = wave_sum32(hub);
  if (lane == 0) { sB[w] = wb; sH[w] = wh; }
  __syncthreads();
  if (tid == 0) {
    float tb = 0.0f, th = 0.0f;
    for (int k = 0; k < 8; ++k) { tb += sB[k]; th += sH[k]; }
    out[0] = tb / (float)(posV + negV);          // cls_loss
    out[1] = th / ((float)posV * 16.0f);         // (huber / (posV*4)) / 4
  }
}

// ---------------------------------------------------------------------------
extern "C" void kernel_launch(void* const* d_in, const int* in_sizes, int n_in,
                              void* d_out, int out_size, void* d_ws, size_t ws_size,
                              hipStream_t stream) {
  (void)in_sizes; (void)n_in; (void)out_size; (void)ws_size;
  const float* reg     = (const float*)d_in[0];   // (B, A, 4)
  const float* cls     = (const float*)d_in[1];   // (B, A)
  const float* anchors = (const float*)d_in[2];   // (A, 4)
  const float* targets = (const float*)d_in[3];   // (B, T, 4)
  float* out = (float*)d_out;                     // [cls_loss, reg_loss/4]

  int* cnt    = (int*)d_ws;                       // cnt[0] = n_pos
  int* posIdx = cnt + 4;                          // 128 ints
  int* negIdx = posIdx + POS_CAP;                 // 248 ints
  unsigned char* mb = (unsigned char*)(negIdx + NEG_CAP);  // TOT_N bytes

  // zero counter + index tables each call (filler slots must gather index 0)
  (void)hipMemsetAsync(d_ws, 0, (4 + POS_CAP + NEG_CAP) * sizeof(int), stream);

  dim3 g1(A_N / 256, B_N);
  iou_kernel<<<g1, 256, 0, stream>>>(anchors, targets, mb, cnt);
  select_kernel<<<1, 1024, 0, stream>>>(mb, cnt, posIdx, negIdx);
  loss_kernel<<<1, 256, 0, stream>>>(reg, cls, anchors, targets, mb,
                                     posIdx, negIdx, cnt, out);
}